// Net_30339648979593
// MI455X (gfx1250) — compile-verified
//
#include <hip/hip_runtime.h>
#include <math.h>

typedef __attribute__((ext_vector_type(2))) float v2f;
typedef __attribute__((ext_vector_type(8))) float v8f;

#define N_NODES 100000
#define N_EDGES 1600000
#define NFEAT   128
#define NHID    64
#define NCLASS  40
#define NPAD    48   // NCLASS padded to 3 WMMA tiles

// ---------------- degree / normalization ----------------

__global__ void k_init_deg(int* deg) {
    int i = blockIdx.x * blockDim.x + threadIdx.x;
    if (i < N_NODES) deg[i] = 1;            // self loop
}

__global__ void k_count_deg(const int* __restrict__ src, int* deg) {
    int e = blockIdx.x * blockDim.x + threadIdx.x;
    if (e < N_EDGES) atomicAdd(&deg[src[e]], 1);
}

__global__ void k_dinv(const int* __restrict__ deg, float* __restrict__ dinv) {
    int i = blockIdx.x * blockDim.x + threadIdx.x;
    if (i < N_NODES) dinv[i] = rsqrtf((float)deg[i]);
}

__global__ void k_norm(const int* __restrict__ src, const int* __restrict__ dst,
                       const float* __restrict__ dinv, float* __restrict__ nrm) {
    int e = blockIdx.x * blockDim.x + threadIdx.x;
    if (e < N_EDGES) nrm[e] = dinv[src[e]] * dinv[dst[e]];
}

// ---------------- GEMM1: t1 = x @ W1^T + b1   [N,128]x[128,64] ----------------
// 256 threads = 8 waves; wave w computes rows [blk*128 + w*16, +16) x all 64 cols
// as 4 WMMA 16x16 tiles, K accumulated in steps of 4.

__global__ __launch_bounds__(256) void k_gemm1(const float* __restrict__ X,
                                               const float* __restrict__ W1,
                                               const float* __restrict__ B1,
                                               float* __restrict__ T1) {
    __shared__ float w[NHID * NFEAT];        // 32 KB, row-major [64][128]
    {
        const float4* s4 = (const float4*)W1;
        float4* d4 = (float4*)w;
        for (int i = threadIdx.x; i < NHID * NFEAT / 4; i += 256) d4[i] = s4[i];
    }
    __syncthreads();

    const int wave = threadIdx.x >> 5;
    const int lane = threadIdx.x & 31;
    const int r    = lane & 15;              // col-within-tile / row-within-strip
    const int kg   = lane >> 4;              // K-pair group (0 or 1)
    const int row0 = blockIdx.x * 128 + wave * 16;

    int rowA = row0 + r;
    if (rowA >= N_NODES) rowA = N_NODES - 1; // clamp (stores are guarded)
    const float* xrow = X + (size_t)rowA * NFEAT;

    v8f acc0 = {}, acc1 = {}, acc2 = {}, acc3 = {};

    for (int k = 0; k < NFEAT; k += 4) {
        const int kk = k + kg * 2;
        v2f a;  a.x = xrow[kk];                 a.y = xrow[kk + 1];
        v2f b0; b0.x = w[(r     ) * NFEAT + kk]; b0.y = w[(r     ) * NFEAT + kk + 1];
        v2f b1; b1.x = w[(r + 16) * NFEAT + kk]; b1.y = w[(r + 16) * NFEAT + kk + 1];
        v2f b2; b2.x = w[(r + 32) * NFEAT + kk]; b2.y = w[(r + 32) * NFEAT + kk + 1];
        v2f b3; b3.x = w[(r + 48) * NFEAT + kk]; b3.y = w[(r + 48) * NFEAT + kk + 1];
        acc0 = __builtin_amdgcn_wmma_f32_16x16x4_f32(false, a, false, b0, (short)0, acc0, false, false);
        acc1 = __builtin_amdgcn_wmma_f32_16x16x4_f32(false, a, false, b1, (short)0, acc1, false, false);
        acc2 = __builtin_amdgcn_wmma_f32_16x16x4_f32(false, a, false, b2, (short)0, acc2, false, false);
        acc3 = __builtin_amdgcn_wmma_f32_16x16x4_f32(false, a, false, b3, (short)0, acc3, false, false);
    }

    const float bb0 = B1[r], bb1 = B1[r + 16], bb2 = B1[r + 32], bb3 = B1[r + 48];
#pragma unroll
    for (int v = 0; v < 8; ++v) {
        const int m = row0 + kg * 8 + v;     // C/D layout: m = 8*(lane>=16) + vgpr
        if (m < N_NODES) {
            float* o = T1 + (size_t)m * NHID;
            o[r     ] = acc0[v] + bb0;
            o[r + 16] = acc1[v] + bb1;
            o[r + 32] = acc2[v] + bb2;
            o[r + 48] = acc3[v] + bb3;
        }
    }
}

// ---------------- aggregation layer 1 ----------------

__global__ void k_selfloop1(const float* __restrict__ dinv,
                            const float* __restrict__ t1, float* __restrict__ t2) {
    int gid = blockIdx.x * blockDim.x + threadIdx.x;
    if (gid < N_NODES * NHID) {
        float di = dinv[gid >> 6];           // NHID == 64
        t2[gid] = di * di * t1[gid];
    }
}

__global__ void k_agg1(const int* __restrict__ src, const int* __restrict__ dst,
                       const float* __restrict__ nrm, const float* __restrict__ t1,
                       float* __restrict__ t2) {
    long long gid = (long long)blockIdx.x * blockDim.x + threadIdx.x;
    int e = (int)(gid >> 6);
    int f = (int)(gid & 63);
    if (e < N_EDGES) {
        int s = src[e], d = dst[e];
        float wv = nrm[e] * t1[(size_t)s * NHID + f];
        atomicAdd(&t2[(size_t)d * NHID + f], wv);
    }
}

// ---------------- GEMM2: t3 = relu(t2) @ W2^T + b2   [N,64]x[64,40] ----------------

__global__ __launch_bounds__(256) void k_gemm2(const float* __restrict__ T2,
                                               const float* __restrict__ W2,
                                               const float* __restrict__ B2,
                                               float* __restrict__ T3) {
    __shared__ float w[NPAD * NHID];         // 12 KB, [48][64], rows 40..47 zero
    for (int i = threadIdx.x; i < NPAD * NHID; i += 256) {
        int o = i >> 6, kk = i & 63;
        w[i] = (o < NCLASS) ? W2[o * NHID + kk] : 0.0f;
    }
    __syncthreads();

    const int wave = threadIdx.x >> 5;
    const int lane = threadIdx.x & 31;
    const int r    = lane & 15;
    const int kg   = lane >> 4;
    const int row0 = blockIdx.x * 128 + wave * 16;

    int rowA = row0 + r;
    if (rowA >= N_NODES) rowA = N_NODES - 1;
    const float* arow = T2 + (size_t)rowA * NHID;

    v8f acc0 = {}, acc1 = {}, acc2 = {};

    for (int k = 0; k < NHID; k += 4) {
        const int kk = k + kg * 2;
        v2f a;  a.x = fmaxf(arow[kk], 0.0f);  a.y = fmaxf(arow[kk + 1], 0.0f); // fused ReLU
        v2f b0; b0.x = w[(r     ) * NHID + kk]; b0.y = w[(r     ) * NHID + kk + 1];
        v2f b1; b1.x = w[(r + 16) * NHID + kk]; b1.y = w[(r + 16) * NHID + kk + 1];
        v2f b2; b2.x = w[(r + 32) * NHID + kk]; b2.y = w[(r + 32) * NHID + kk + 1];
        acc0 = __builtin_amdgcn_wmma_f32_16x16x4_f32(false, a, false, b0, (short)0, acc0, false, false);
        acc1 = __builtin_amdgcn_wmma_f32_16x16x4_f32(false, a, false, b1, (short)0, acc1, false, false);
        acc2 = __builtin_amdgcn_wmma_f32_16x16x4_f32(false, a, false, b2, (short)0, acc2, false, false);
    }

    const float bb0 = B2[r], bb1 = B2[r + 16], bb2 = (r + 32 < NCLASS) ? B2[r + 32] : 0.0f;
#pragma unroll
    for (int v = 0; v < 8; ++v) {
        const int m = row0 + kg * 8 + v;
        if (m < N_NODES) {
            float* o = T3 + (size_t)m * NCLASS;
            o[r     ] = acc0[v] + bb0;
            o[r + 16] = acc1[v] + bb1;
            if (r + 32 < NCLASS) o[r + 32] = acc2[v] + bb2;
        }
    }
}

// ---------------- aggregation layer 2 (into d_out) ----------------

__global__ void k_selfloop2(const float* __restrict__ dinv,
                            const float* __restrict__ t3, float* __restrict__ out) {
    int gid = blockIdx.x * blockDim.x + threadIdx.x;
    if (gid < N_NODES * NCLASS) {
        float di = dinv[gid / NCLASS];
        out[gid] = di * di * t3[gid];
    }
}

__global__ void k_agg2(const int* __restrict__ src, const int* __restrict__ dst,
                       const float* __restrict__ nrm, const float* __restrict__ t3,
                       float* __restrict__ out) {
    long long gid = (long long)blockIdx.x * blockDim.x + threadIdx.x;
    int e = (int)(gid >> 6);
    int f = (int)(gid & 63);
    if (e < N_EDGES && f < NCLASS) {
        int s = src[e], d = dst[e];
        float wv = nrm[e] * t3[(size_t)s * NCLASS + f];
        atomicAdd(&out[(size_t)d * NCLASS + f], wv);
    }
}

// ---------------- log_softmax, in place, one wave per row ----------------

__global__ __launch_bounds__(256) void k_logsoftmax(float* __restrict__ out) {
    int row  = blockIdx.x * 8 + (threadIdx.x >> 5);
    int lane = threadIdx.x & 31;
    if (row >= N_NODES) return;
    float* p = out + (size_t)row * NCLASS;

    float v0 = p[lane];
    float v1 = (lane < NCLASS - 32) ? p[lane + 32] : -INFINITY;
    float m = fmaxf(v0, v1);
#pragma unroll
    for (int off = 16; off > 0; off >>= 1) m = fmaxf(m, __shfl_xor(m, off, 32));
    float s = expf(v0 - m) + ((lane < NCLASS - 32) ? expf(v1 - m) : 0.0f);
#pragma unroll
    for (int off = 16; off > 0; off >>= 1) s += __shfl_xor(s, off, 32);
    float ls = logf(s);
    p[lane] = v0 - m - ls;
    if (lane < NCLASS - 32) p[lane + 32] = v1 - m - ls;
}

// ---------------- launcher ----------------

extern "C" void kernel_launch(void* const* d_in, const int* in_sizes, int n_in,
                              void* d_out, int out_size, void* d_ws, size_t ws_size,
                              hipStream_t stream) {
    (void)in_sizes; (void)n_in; (void)out_size; (void)ws_size;
    const float* x  = (const float*)d_in[0];
    const float* W1 = (const float*)d_in[1];
    const float* b1 = (const float*)d_in[2];
    const float* W2 = (const float*)d_in[3];
    const float* b2 = (const float*)d_in[4];
    const int*   ei = (const int*)d_in[5];
    const int* src = ei;
    const int* dst = ei + N_EDGES;
    float* out = (float*)d_out;

    char* ws = (char*)d_ws;
    size_t off = 0;
    auto carve = [&](size_t bytes) -> void* {
        void* p = ws + off;
        off += (bytes + 255) & ~(size_t)255;
        return p;
    };
    int*   deg  = (int*)  carve((size_t)N_NODES * 4);
    float* dinv = (float*)carve((size_t)N_NODES * 4);
    float* nrm  = (float*)carve((size_t)N_EDGES * 4);
    float* t1   = (float*)carve((size_t)N_NODES * NHID * 4);
    float* t2   = (float*)carve((size_t)N_NODES * NHID * 4);
    float* t3   = (float*)carve((size_t)N_NODES * NCLASS * 4);

    const int B = 256;
    int gN  = (N_NODES + B - 1) / B;
    int gE  = (N_EDGES + B - 1) / B;
    int gG  = (N_NODES + 127) / 128;                      // 782
    int gA  = (int)(((long long)N_EDGES * 64 + B - 1) / B);
    int gS1 = (N_NODES * NHID + B - 1) / B;
    int gS2 = (N_NODES * NCLASS + B - 1) / B;
    int gLS = (N_NODES + 7) / 8;

    k_init_deg  <<<gN,  B, 0, stream>>>(deg);
    k_count_deg <<<gE,  B, 0, stream>>>(src, deg);
    k_dinv      <<<gN,  B, 0, stream>>>(deg, dinv);
    k_norm      <<<gE,  B, 0, stream>>>(src, dst, dinv, nrm);

    k_gemm1     <<<gG,  B, 0, stream>>>(x, W1, b1, t1);
    k_selfloop1 <<<gS1, B, 0, stream>>>(dinv, t1, t2);
    k_agg1      <<<gA,  B, 0, stream>>>(src, dst, nrm, t1, t2);

    k_gemm2     <<<gG,  B, 0, stream>>>(t2, W2, b2, t3);
    k_selfloop2 <<<gS2, B, 0, stream>>>(dinv, t3, out);
    k_agg2      <<<gA,  B, 0, stream>>>(src, dst, nrm, t3, out);

    k_logsoftmax<<<gLS, B, 0, stream>>>(out);
}